// HyperGIN_57638461112981
// MI455X (gfx1250) — compile-verified
//
#include <hip/hip_runtime.h>
#include <math.h>

// ---------------- problem constants ----------------
constexpr int kN  = 50000;   // nodes
constexpr int kE  = 800000;  // edges
constexpr int kIN = 200;     // input features
constexpr int kH  = 128;     // hyperboloid dim (time + 127 spatial)
constexpr int kD  = 127;     // spatial dim
constexpr float kEPS = 1e-5f;

typedef float v2f __attribute__((ext_vector_type(2)));
typedef float v8f __attribute__((ext_vector_type(8)));

// ---------------- helpers ----------------
__device__ inline float wave_sum(float v) {
#pragma unroll
  for (int off = 16; off > 0; off >>= 1) v += __shfl_xor(v, off, 32);
  return v;
}

// ---------------- zero fill ----------------
__global__ void k_zero(float* __restrict__ p, size_t n) {
  size_t i = (size_t)blockIdx.x * blockDim.x + threadIdx.x;
  size_t stride = (size_t)gridDim.x * blockDim.x;
  for (; i < n; i += stride) p[i] = 0.0f;
}

// ---------------- f32 WMMA GEMM:  C[n,j] = sum_k A[n,k] * B(k,j) (+bias[j]) ----
// transB==0 : B(k,j) = B[k*ldb + j]   (input embed, w_in is (K, D))
// transB==1 : B(k,j) = B[j*ldb + k]   (hyp_linear uses w.T, w is (D, D))
// block = 256 threads = 8 waves; block tile = 64 rows x 128 cols;
// wave tile = 16 rows x 64 cols = 4 x V_WMMA_F32_16X16X4_F32 accumulators.
// B is staged through a 64 KB LDS tile (128 k x 128 col, k-pair interleaved so
// each lane reads {B[k][col], B[k+1][col]} as one aligned ds_load_b64).
// Requirements: Kpad % 4 == 0 and A rows readable for kk < Kpad (padded cols
// of the logmap buffers are written as 0, so they contribute nothing).
#define KCHUNK 128

__global__ __launch_bounds__(256)
void k_gemm(const float* __restrict__ A, int lda, int nrows, int K, int Kpad,
            const float* __restrict__ B, int ldb, int transB, int ncol,
            const float* __restrict__ bias,
            float* __restrict__ C, int ldc) {
  __shared__ float Bs[KCHUNK * 128];     // pair-interleaved: [(k>>1)*128+col]*2+(k&1)

  const int lane = threadIdx.x & 31;
  const int wave = threadIdx.x >> 5;
  const int rowBase = blockIdx.x * 64 + (wave >> 1) * 16;
  const int colBase = (wave & 1) * 64;

  const int m  = lane & 15;              // A row within tile / C col lane
  const int hi = lane >> 4;              // half-wave select
  int arow = rowBase + m;
  if (arow >= nrows) arow = nrows - 1;   // clamp read; store is guarded
  const float* Ap = A + (size_t)arow * lda;

  v8f acc[4] = {v8f{}, v8f{}, v8f{}, v8f{}};

  for (int k0 = 0; k0 < Kpad; k0 += KCHUNK) {
    const int kc = (Kpad - k0 < KCHUNK) ? (Kpad - k0) : KCHUNK;

    // cooperative fill of the B chunk (guards live here, not in the MMA loop)
    __syncthreads();
    for (int idx = threadIdx.x; idx < kc * 128; idx += 256) {
      const int kl  = idx >> 7;          // local k
      const int col = idx & 127;
      const int kg  = k0 + kl;           // global k
      float v = 0.0f;
      if (kg < K && col < ncol)
        v = transB ? B[(size_t)col * ldb + kg] : B[(size_t)kg * ldb + col];
      Bs[((kl >> 1) * 128 + col) * 2 + (kl & 1)] = v;
    }
    __syncthreads();

    // branch-free MMA loop: 1x global_load_b64 (A) + 4x ds_load_b64 (B) + 4x WMMA
    for (int k = 0; k < kc; k += 4) {
      const int kk = k + (hi << 1);      // local, even
      const float2 af = *(const float2*)(Ap + (k0 + kk));
      v2f a; a.x = af.x; a.y = af.y;
#pragma unroll
      for (int t = 0; t < 4; ++t) {
        const int col = colBase + t * 16 + m;
        const float2 bf = *(const float2*)(Bs + ((kk >> 1) * 128 + col) * 2);
        v2f b; b.x = bf.x; b.y = bf.y;
        acc[t] = __builtin_amdgcn_wmma_f32_16x16x4_f32(
            false, a, false, b, (short)0, acc[t], false, false);
      }
    }
  }

#pragma unroll
  for (int t = 0; t < 4; ++t) {
    const int col = colBase + t * 16 + m;
    const float bv = (bias && col < ncol) ? bias[col] : 0.0f;
#pragma unroll
    for (int r = 0; r < 8; ++r) {
      const int row = rowBase + r + (hi << 3);
      if (row < nrows && col < ncol)
        C[(size_t)row * ldc + col] = acc[t][r] + bv;
    }
  }
}

// ---------------- proj(expmap0([*, u])) : spatial u in g[0..126] -> full row ----
__global__ __launch_bounds__(256)
void k_expmap_proj(const float* __restrict__ gin, float* __restrict__ hout,
                   const float* __restrict__ cptr, int nrows) {
  const int wid = blockIdx.x * 8 + (threadIdx.x >> 5);
  if (wid >= nrows) return;
  const int lane = threadIdx.x & 31;
  const float c = cptr[0], sc = sqrtf(c);
  const float* g = gin + (size_t)wid * kH;
  float u[4], ss = 0.0f;
#pragma unroll
  for (int i = 0; i < 4; ++i) {
    const int j = lane * 4 + i;
    const float v = (j < kD) ? g[j] : 0.0f;
    u[i] = v; ss += v * v;
  }
  const float nn = fmaxf(sqrtf(wave_sum(ss)), 1e-9f);
  const float sh = sinhf(nn / sc);
  const float coef = sc * sh / nn;
  const float time = sqrtf(c + c * sh * sh);   // proj recomputed time
  float* o = hout + (size_t)wid * kH;
#pragma unroll
  for (int i = 0; i < 4; ++i) {
    const int j = lane * 4 + i;
    if (j + 1 < kH) o[j + 1] = coef * u[i];
  }
  if (lane == 0) o[0] = time;
}

// ---------------- logmap0: full row x -> spatial result at cols 0..126 (127 = 0)
__global__ __launch_bounds__(256)
void k_logmap(const float* __restrict__ xin, float* __restrict__ vout,
              const float* __restrict__ cptr, int nrows) {
  const int wid = blockIdx.x * 8 + (threadIdx.x >> 5);
  if (wid >= nrows) return;
  const int lane = threadIdx.x & 31;
  const float c = cptr[0], sc = sqrtf(c);
  const float* x = xin + (size_t)wid * kH;
  float y[4], ss = 0.0f;
#pragma unroll
  for (int i = 0; i < 4; ++i) {
    const int j = lane * 4 + i;
    const float v = (j < kD) ? x[1 + j] : 0.0f;
    y[i] = v; ss += v * v;
  }
  const float yn = fmaxf(sqrtf(wave_sum(ss)), 1e-9f);
  const float x0 = x[0];
  const float d = sc * acoshf(fmaxf(x0 / sc, 1.0f + kEPS));
  const float coef = d / yn;
  float* o = vout + (size_t)wid * kH;
#pragma unroll
  for (int i = 0; i < 4; ++i) {
    const int j = lane * 4 + i;
    o[j] = (j < kD) ? coef * y[i] : 0.0f;    // col 127 written as 0 (GEMM pad)
  }
}

// ---------------- att[e] = exp(-dist(hl[src], hl[dst])) ----------------
__global__ __launch_bounds__(256)
void k_edge_att(const float* __restrict__ hl, const int* __restrict__ ei,
                const float* __restrict__ cptr, float* __restrict__ att, int ne) {
  const int e = blockIdx.x * 8 + (threadIdx.x >> 5);
  if (e >= ne) return;
  const int lane = threadIdx.x & 31;
  const int s = ei[e], d = ei[ne + e];
  const float4 xv = ((const float4*)(hl + (size_t)s * kH))[lane];
  const float4 yv = ((const float4*)(hl + (size_t)d * kH))[lane];
  float t = xv.x * yv.x + xv.y * yv.y + xv.z * yv.z + xv.w * yv.w;
  if (lane == 0) t -= 2.0f * xv.x * yv.x;      // Lorentz sign on time comp
  const float inner = wave_sum(t);
  if (lane == 0) {
    const float c = cptr[0];
    const float val = fmaxf(-inner / c, 1.0f + kEPS);
    att[e] = expf(-sqrtf(c) * acoshf(val));
  }
}

// ---------------- m[src] += att[e] * h[dst] (f32 atomics, L2-resident) --------
__global__ __launch_bounds__(256)
void k_scatter(const float* __restrict__ att, const float* __restrict__ h,
               const int* __restrict__ ei, float* __restrict__ mbuf, int ne) {
  const int e = blockIdx.x * 8 + (threadIdx.x >> 5);
  if (e >= ne) return;
  const int lane = threadIdx.x & 31;
  const float a = att[e];
  const int s = ei[e], d = ei[ne + e];
  const float4 hv = ((const float4*)(h + (size_t)d * kH))[lane];
  float* mp = mbuf + (size_t)s * kH + lane * 4;
  atomicAdd(mp + 0, a * hv.x);
  atomicAdd(mp + 1, a * hv.y);
  atomicAdd(mp + 2, a * hv.z);
  atomicAdd(mp + 3, a * hv.w);
}

// ---------------- agg = proj( sqrt(c) * m / sqrt(max(-<m,m>_L, 1e-9)) ) -------
__global__ __launch_bounds__(256)
void k_centroid(float* __restrict__ mbuf, const float* __restrict__ cptr, int nrows) {
  const int wid = blockIdx.x * 8 + (threadIdx.x >> 5);
  if (wid >= nrows) return;
  const int lane = threadIdx.x & 31;
  float* row = mbuf + (size_t)wid * kH;
  float4 mv = ((const float4*)row)[lane];
  float t = mv.x * mv.x + mv.y * mv.y + mv.z * mv.z + mv.w * mv.w;
  if (lane == 0) t -= mv.x * mv.x;                 // spatial-only squares
  const float sum_sp = wave_sum(t);
  const float m0 = __shfl(mv.x, 0, 32);
  const float c = cptr[0], sc = sqrtf(c);
  const float denom = sqrtf(fmaxf(m0 * m0 - sum_sp, 1e-9f));
  const float coef = sc / denom;
  const float time = sqrtf(c + coef * coef * sum_sp);  // proj time
  float4 o;
  o.x = (lane == 0) ? time : coef * mv.x;
  o.y = coef * mv.y; o.z = coef * mv.z; o.w = coef * mv.w;
  ((float4*)row)[lane] = o;
}

// ------ h' = proj(expmap0( logmap0(agg) + (1+eps)*logmap0(proj(h)) )) ---------
__global__ __launch_bounds__(256)
void k_combine(const float* __restrict__ agg, float* __restrict__ h,
               const float* __restrict__ cptr, const float* __restrict__ epsp,
               int layer, int nrows) {
  const int wid = blockIdx.x * 8 + (threadIdx.x >> 5);
  if (wid >= nrows) return;
  const int lane = threadIdx.x & 31;
  const float c = cptr[0], sc = sqrtf(c);
  const float ep = 1.0f + epsp[layer];
  const float4 av = ((const float4*)(agg + (size_t)wid * kH))[lane];
  float* hrow = h + (size_t)wid * kH;
  const float4 hv = ((const float4*)hrow)[lane];

  float ta = av.x * av.x + av.y * av.y + av.z * av.z + av.w * av.w;
  float th = hv.x * hv.x + hv.y * hv.y + hv.z * hv.z + hv.w * hv.w;
  if (lane == 0) { ta -= av.x * av.x; th -= hv.x * hv.x; }
  const float sa = wave_sum(ta);          // ||agg_sp||^2
  const float sh = wave_sum(th);          // ||h_sp||^2
  const float a0 = __shfl(av.x, 0, 32);

  const float da = sc * acoshf(fmaxf(a0 / sc, 1.0f + kEPS));
  const float ca = da / fmaxf(sqrtf(sa), 1e-9f);
  const float h0p = sqrtf(c + sh);        // proj(h) time
  const float dh = sc * acoshf(fmaxf(h0p / sc, 1.0f + kEPS));
  const float ch = ep * dh / fmaxf(sqrtf(sh), 1e-9f);

  float vx = ca * av.x + ch * hv.x;       // time comp of v is 0:
  if (lane == 0) vx = 0.0f;
  const float vy = ca * av.y + ch * hv.y;
  const float vz = ca * av.z + ch * hv.z;
  const float vw = ca * av.w + ch * hv.w;

  const float n2 = wave_sum(vx * vx + vy * vy + vz * vz + vw * vw);
  const float nn = fmaxf(sqrtf(n2), 1e-9f);
  const float shh = sinhf(nn / sc);
  const float coef = sc * shh / nn;
  const float time = sqrtf(c + c * shh * shh);
  float4 o;
  o.x = (lane == 0) ? time : coef * vx;
  o.y = coef * vy; o.z = coef * vz; o.w = coef * vw;
  ((float4*)hrow)[lane] = o;
}

// ---------------- hyp_act: expmap0([0, tanh(logmap0(x)_sp)]) ------------------
__global__ __launch_bounds__(256)
void k_act(const float* __restrict__ xin, float* __restrict__ hout,
           const float* __restrict__ cptr, int nrows) {
  const int wid = blockIdx.x * 8 + (threadIdx.x >> 5);
  if (wid >= nrows) return;
  const int lane = threadIdx.x & 31;
  const float c = cptr[0], sc = sqrtf(c);
  const float4 xv = ((const float4*)(xin + (size_t)wid * kH))[lane];
  float t = xv.x * xv.x + xv.y * xv.y + xv.z * xv.z + xv.w * xv.w;
  if (lane == 0) t -= xv.x * xv.x;
  const float sum_sp = wave_sum(t);
  const float x0 = __shfl(xv.x, 0, 32);
  const float d = sc * acoshf(fmaxf(x0 / sc, 1.0f + kEPS));
  const float cl = d / fmaxf(sqrtf(sum_sp), 1e-9f);
  float vx = (lane == 0) ? 0.0f : tanhf(cl * xv.x);
  const float vy = tanhf(cl * xv.y);
  const float vz = tanhf(cl * xv.z);
  const float vw = tanhf(cl * xv.w);
  const float n2 = wave_sum(vx * vx + vy * vy + vz * vz + vw * vw);
  const float nn = fmaxf(sqrtf(n2), 1e-9f);
  const float coef = sc * sinhf(nn / sc) / nn;
  const float time = sc * coshf(nn / sc);        // no proj (matches reference)
  float4 o;
  o.x = (lane == 0) ? time : coef * vx;
  o.y = coef * vy; o.z = coef * vz; o.w = coef * vw;
  ((float4*)(hout + (size_t)wid * kH))[lane] = o;
}

// ---------------- pooled[l] = segment_sum(logmap0(h), batch) ------------------
__global__ __launch_bounds__(256)
void k_pool(const float* __restrict__ h, const int* __restrict__ batch,
            const float* __restrict__ cptr, float* __restrict__ out,
            int layer, int nrows) {
  const int wid = blockIdx.x * 8 + (threadIdx.x >> 5);
  if (wid >= nrows) return;
  const int lane = threadIdx.x & 31;
  const float c = cptr[0], sc = sqrtf(c);
  const float* x = h + (size_t)wid * kH;
  float y[4], ss = 0.0f;
#pragma unroll
  for (int i = 0; i < 4; ++i) {
    const int j = lane * 4 + i;
    const float v = (j < kD) ? x[1 + j] : 0.0f;
    y[i] = v; ss += v * v;
  }
  const float yn = fmaxf(sqrtf(wave_sum(ss)), 1e-9f);
  const float d = sc * acoshf(fmaxf(x[0] / sc, 1.0f + kEPS));
  const float coef = d / yn;
  const int b = batch[wid];
  float* orow = out + (size_t)b * (2 * kH) + (size_t)layer * kH;
#pragma unroll
  for (int i = 0; i < 4; ++i) {
    const int j = lane * 4 + i;
    if (j < kD) atomicAdd(&orow[1 + j], coef * y[i]);  // time col stays 0
  }
}

// ---------------- host orchestration ----------------
extern "C" void kernel_launch(void* const* d_in, const int* in_sizes, int n_in,
                              void* d_out, int out_size, void* d_ws, size_t ws_size,
                              hipStream_t stream) {
  const float* x     = (const float*)d_in[0];
  const float* c     = (const float*)d_in[1];
  const float* w_in  = (const float*)d_in[2];
  const float* att_w = (const float*)d_in[3];
  const float* att_b = (const float*)d_in[4];
  const float* eps   = (const float*)d_in[5];
  const float* mlp_w = (const float*)d_in[6];
  const float* mlp_b = (const float*)d_in[7];
  const int*   ei    = (const int*)d_in[8];
  const int*   batch = (const int*)d_in[9];
  float* out = (float*)d_out;

  float* ws = (float*)d_ws;
  const size_t NP = (size_t)kN * kH;          // 6.4M floats = 25.6 MB
  float* bufH  = ws;                          // current node states h
  float* bufV  = ws + 1 * NP;                 // logmap scratch (GEMM A, col127=0)
  float* bufG  = ws + 2 * NP;                 // GEMM output (spatial)
  float* bufHL = ws + 3 * NP;                 // hl / linear results
  float* bufM  = ws + 4 * NP;                 // centroid accumulator
  float* att   = ws + 5 * NP;                 // per-edge attention (E floats)

  const dim3 blk(256);
  const int nodeBlocks = (kN + 7) / 8;
  const int edgeBlocks = (kE + 7) / 8;
  const int gemmBlocks = (kN + 63) / 64;

  k_zero<<<256, blk, 0, stream>>>(out, (size_t)out_size);

  // input embed: v0_sp = x @ w_in (K=200, no pad needed), then h = proj(expmap0)
  k_gemm<<<gemmBlocks, blk, 0, stream>>>(x, kIN, kN, kIN, kIN, w_in, kD, 0, kD,
                                         nullptr, bufG, kH);
  k_expmap_proj<<<nodeBlocks, blk, 0, stream>>>(bufG, bufH, c, kN);

  for (int l = 0; l < 2; ++l) {
    // hl = hyp_linear(h, att_w[l], att_b[l])   (K=127, Kpad=128 via zero col)
    k_logmap<<<nodeBlocks, blk, 0, stream>>>(bufH, bufV, c, kN);
    k_gemm<<<gemmBlocks, blk, 0, stream>>>(bufV, kH, kN, kD, kH,
                                           att_w + (size_t)l * kD * kD, kD, 1, kD,
                                           att_b + (size_t)l * kD, bufG, kH);
    k_expmap_proj<<<nodeBlocks, blk, 0, stream>>>(bufG, bufHL, c, kN);

    // edge attention + centroid aggregation
    k_edge_att<<<edgeBlocks, blk, 0, stream>>>(bufHL, ei, c, att, kE);
    k_zero<<<256, blk, 0, stream>>>(bufM, NP);
    k_scatter<<<edgeBlocks, blk, 0, stream>>>(att, bufH, ei, bufM, kE);
    k_centroid<<<nodeBlocks, blk, 0, stream>>>(bufM, c, kN);

    // h = proj(expmap0(logmap0(agg) + (1+eps)*logmap0(proj(h))))
    k_combine<<<nodeBlocks, blk, 0, stream>>>(bufM, bufH, c, eps, l, kN);

    // MLP: 2 x hyp_act(hyp_linear(...))
    for (int m = 0; m < 2; ++m) {
      k_logmap<<<nodeBlocks, blk, 0, stream>>>(bufH, bufV, c, kN);
      k_gemm<<<gemmBlocks, blk, 0, stream>>>(bufV, kH, kN, kD, kH,
                                             mlp_w + (size_t)(l * 2 + m) * kD * kD, kD, 1, kD,
                                             mlp_b + (size_t)(l * 2 + m) * kD, bufG, kH);
      k_expmap_proj<<<nodeBlocks, blk, 0, stream>>>(bufG, bufHL, c, kN);
      k_act<<<nodeBlocks, blk, 0, stream>>>(bufHL, bufH, c, kN);
    }

    // pooled[l] = segment_sum(logmap0(h), batch)
    k_pool<<<nodeBlocks, blk, 0, stream>>>(bufH, batch, c, out, l, kN);
  }
}